// Router_23983097381582
// MI455X (gfx1250) — compile-verified
//
#include <hip/hip_runtime.h>
#include <math.h>

// Problem constants (from reference): B=64, C=512, H=W=32 (HW=1024), N=1, A=16
#define BB   64
#define CC   512
#define HW   1024
#define AA   16
#define EPSF 1e-5f

typedef float v2f __attribute__((ext_vector_type(2)));
typedef float v8f __attribute__((ext_vector_type(8)));

// ---------------------------------------------------------------------------
// Kernel 1: streaming reduction over x. One wave per (b,c) pair (contiguous
// 1024 floats = 4KB). Each lane grabs 8 x float4 (fully coalesced 512B/wave
// chunks), then butterfly-reduces sum and sum-of-squares across the wave32.
// This is the only kernel that touches the 128MB tensor -> HBM-bound.
// ---------------------------------------------------------------------------
__global__ void k_reduce_x(const float* __restrict__ x,
                           float* __restrict__ S, float* __restrict__ SS) {
  int wave = blockIdx.x * (blockDim.x >> 5) + (threadIdx.x >> 5); // (b*C + c)
  int lane = threadIdx.x & 31;
  const float4* base = (const float4*)(x + (size_t)wave * HW);
  float s = 0.f, q = 0.f;
#pragma unroll
  for (int i = 0; i < 8; ++i) {
    float4 v = base[i * 32 + lane];
    s += (v.x + v.y) + (v.z + v.w);
    q += (v.x * v.x + v.y * v.y) + (v.z * v.z + v.w * v.w);
  }
#pragma unroll
  for (int m = 16; m >= 1; m >>= 1) {
    s += __shfl_xor(s, m, 32);
    q += __shfl_xor(q, m, 32);
  }
  if (lane == 0) { S[wave] = s; SS[wave] = q; }
}

// ---------------------------------------------------------------------------
// Kernel 2: per-channel batch stats. Thread c sums S/SS over b (coalesced:
// consecutive threads read consecutive floats of each 512-float row).
// ---------------------------------------------------------------------------
__global__ void k_stats(const float* __restrict__ S, const float* __restrict__ SS,
                        float* __restrict__ mu, float* __restrict__ rstd) {
  int c = blockIdx.x * blockDim.x + threadIdx.x;
  if (c >= CC) return;
  float s = 0.f, q = 0.f;
  for (int b = 0; b < BB; ++b) { s += S[b * CC + c]; q += SS[b * CC + c]; }
  const float inv = 1.0f / (float)(BB * HW);
  float m = s * inv;
  float v = q * inv - m * m;                 // biased variance (torch BN norm)
  mu[c]   = m;
  rstd[c] = rsqrtf(v + EPSF);
}

// ---------------------------------------------------------------------------
// Kernel 3: new_x[b,c] = g[c]*(mean_hw(x) - mu[c])*rstd[c] + beta[c]
// (pool-of-affine == affine-of-pool). Written straight into d_out region 0.
// ---------------------------------------------------------------------------
__global__ void k_newx(const float* __restrict__ S, const float* __restrict__ mu,
                       const float* __restrict__ rstd, const float* __restrict__ g,
                       const float* __restrict__ be, float* __restrict__ out) {
  int i = blockIdx.x * blockDim.x + threadIdx.x;
  if (i >= BB * CC) return;
  int c = i & (CC - 1);
  float pooled = S[i] * (1.0f / (float)HW);
  out[i] = g[c] * (pooled - mu[c]) * rstd[c] + be[c];
}

// ---------------------------------------------------------------------------
// Kernel 4: LayerNorm of router weights over C. 16 waves, one per adapter a.
// Lane holds 16 strided channels; butterfly reduce gives mean/var to all lanes.
// wn stored [C][A] (row-major, B-operand order for the GEMM).
// ---------------------------------------------------------------------------
__global__ void k_ln_w(const float* __restrict__ rw, const float* __restrict__ lg,
                       const float* __restrict__ lb, float* __restrict__ wn) {
  int a = threadIdx.x >> 5;        // adapter = wave id (blockDim = 512 -> 16 waves)
  int lane = threadIdx.x & 31;
  float vals[CC / 32];
  float s = 0.f, q = 0.f;
#pragma unroll
  for (int i = 0; i < CC / 32; ++i) {
    int c = lane + i * 32;
    float w = rw[c * AA + a];
    vals[i] = w;
    s += w; q += w * w;
  }
#pragma unroll
  for (int m = 16; m >= 1; m >>= 1) {
    s += __shfl_xor(s, m, 32);
    q += __shfl_xor(q, m, 32);
  }
  float mean = s * (1.0f / (float)CC);
  float var  = q * (1.0f / (float)CC) - mean * mean;
  float rs   = rsqrtf(var + EPSF);
#pragma unroll
  for (int i = 0; i < CC / 32; ++i) {
    int c = lane + i * 32;
    wn[c * AA + a] = (vals[i] - mean) * rs * lg[c] + lb[c];
  }
}

// ---------------------------------------------------------------------------
// Kernel 5: logits = new_x[64,512] @ wn[512,16] + bias; softmax over 16.
// Full-fp32 WMMA: V_WMMA_F32_16X16X4_F32, 4 waves x one 16-row M tile each,
// K stepped by 4 (128 wmma issues per wave). wn staged in LDS (32KB).
// ISA 7.12.2 fragment layouts (wave32):
//   A 16x4 f32 : lane L -> row m=L&15, VGPR v -> K = 2*(L>>4)+v
//   B 4x16 f32 : lane L -> col n=L&15, VGPR v -> K = 2*(L>>4)+v
//   C/D 16x16  : lane L -> col n=L&15, VGPR v -> row m = v + 8*(L>>4)
// ---------------------------------------------------------------------------
__global__ void k_route_wmma(const float* __restrict__ newx,
                             const float* __restrict__ wn,
                             const float* __restrict__ bias,
                             float* __restrict__ out_logits,
                             float* __restrict__ out_probs) {
  __shared__ float wnLds[CC * AA];
  for (int i = threadIdx.x; i < CC * AA; i += blockDim.x) wnLds[i] = wn[i];
  __syncthreads();

  int waveId = threadIdx.x >> 5;   // 0..3 -> M tile
  int lane   = threadIdx.x & 31;
  int l15    = lane & 15;
  int half   = lane >> 4;
  int ks     = half * 2;

  const float* arow = newx + (size_t)(waveId * 16 + l15) * CC + ks;

  v8f acc = {0.f, 0.f, 0.f, 0.f, 0.f, 0.f, 0.f, 0.f};
  for (int kk = 0; kk < CC; kk += 4) {
    v2f a = *(const v2f*)(arow + kk);                 // A[m][kk+ks .. kk+ks+1]
    v2f b;
    b.x = wnLds[(kk + ks) * AA + l15];                // B[kk+ks  ][n]
    b.y = wnLds[(kk + ks + 1) * AA + l15];            // B[kk+ks+1][n]
    acc = __builtin_amdgcn_wmma_f32_16x16x4_f32(
        /*neg_a=*/false, a, /*neg_b=*/false, b,
        /*c_mod=*/(short)0, acc, /*reuse_a=*/false, /*reuse_b=*/false);
  }

  float bi = bias[l15];
  float lg[8];
#pragma unroll
  for (int v = 0; v < 8; ++v) {
    int m = v + 8 * half;
    float L = acc[v] + bi;
    lg[v] = L;
    out_logits[(waveId * 16 + m) * AA + l15] = L;
  }
  // Row softmax over n=0..15: shfl_xor 1/2/4/8 stays within each 16-lane half,
  // which is exactly the n-striping of the C/D fragment.
#pragma unroll
  for (int v = 0; v < 8; ++v) {
    float mx = lg[v];
#pragma unroll
    for (int msk = 1; msk <= 8; msk <<= 1) mx = fmaxf(mx, __shfl_xor(mx, msk, 32));
    float e = __expf(lg[v] - mx);
    float sm = e;
#pragma unroll
    for (int msk = 1; msk <= 8; msk <<= 1) sm += __shfl_xor(sm, msk, 32);
    int m = v + 8 * half;
    out_probs[(waveId * 16 + m) * AA + l15] = e / sm;
  }
}

// ---------------------------------------------------------------------------
extern "C" void kernel_launch(void* const* d_in, const int* in_sizes, int n_in,
                              void* d_out, int out_size, void* d_ws, size_t ws_size,
                              hipStream_t stream) {
  (void)in_sizes; (void)n_in; (void)out_size; (void)ws_size;
  const float* x   = (const float*)d_in[0];  // [64,512,32,32]
  const float* rw  = (const float*)d_in[1];  // [1,512,16]
  const float* rb  = (const float*)d_in[2];  // [1,16]
  const float* bng = (const float*)d_in[3];  // [1,512]
  const float* bnb = (const float*)d_in[4];  // [1,512]
  const float* lng = (const float*)d_in[5];  // [1,512]
  const float* lnb = (const float*)d_in[6];  // [1,512]

  float* out  = (float*)d_out;
  float* wsf  = (float*)d_ws;
  float* S    = wsf;                 // 32768
  float* SS   = wsf + 32768;         // 32768
  float* mu   = wsf + 65536;         // 512
  float* rstd = wsf + 66048;         // 512
  float* wn   = wsf + 66560;         // 8192  (total ws: ~292KB)

  float* newx   = out;               // output 0: [1,64,512]
  float* logits = out + BB * CC;     // output 1: [1,64,16]
  float* probs  = logits + BB * AA;  // output 2: [1,64,16]

  // 1) one streaming pass over x (HBM-bound: 128MB)
  k_reduce_x<<<(BB * CC) / 8, 256, 0, stream>>>(x, S, SS);
  // 2) per-channel mu / rstd
  k_stats<<<2, 256, 0, stream>>>(S, SS, mu, rstd);
  // 3) pooled + BN-affine -> new_x (first output region)
  k_newx<<<(BB * CC) / 256, 256, 0, stream>>>(S, mu, rstd, bng, bnb, newx);
  // 4) LayerNorm of router weights -> wn [C][A]
  k_ln_w<<<1, 512, 0, stream>>>(rw, lng, lnb, wn);
  // 5) fp32 WMMA routing GEMM + bias + softmax
  k_route_wmma<<<1, 128, 0, stream>>>(newx, wn, rb, logits, probs);
}